// ContrastiveLoss_58858231824724
// MI455X (gfx1250) — compile-verified
//
#include <hip/hip_runtime.h>

typedef __attribute__((ext_vector_type(2))) float v2f;
typedef __attribute__((ext_vector_type(4))) float v4f;
typedef __attribute__((ext_vector_type(8))) float v8f;

#define CL_N      8192
#define CL_D      512
#define CL_MARGIN 0.3f
#define CL_MTILES (CL_N / 16)            // 512
#define CL_BLOCKS (CL_MTILES / 2)        // 256 paired blocks
#define CL_TOTAL  (CL_MTILES + 1)        // 513 tiles per block pair (balanced)

// ---------------------------------------------------------------------------
// Kernel 1: upper-triangle fused GEMM (V_WMMA_F32_16X16X4_F32) + masked loss.
// Block b owns row-tiles mt1=b and mt2=511-b; work = (512-b)+(b+1)=513 tiles
// for every b -> perfect static load balance. Off-diagonal tiles weighted 2x
// (sim and the masks are symmetric); diagonal tiles weighted 1x.
// ---------------------------------------------------------------------------
__global__ __launch_bounds__(256) void
contrastive_tile_kernel(const float* __restrict__ X,
                        const int*   __restrict__ T,
                        float*       __restrict__ partial)
{
    extern __shared__ float As[];        // 2 A tiles: 2 * 16*512 f32 = 64 KB
    __shared__ int   Tg[32];             // row targets for both tiles
    __shared__ float Red[256];           // block reduction

    const int tid  = threadIdx.x;
    const int lane = tid & 31;
    const int wave = tid >> 5;
    const int half = lane >> 4;          // 0: lanes 0-15, 1: lanes 16-31
    const int l16  = lane & 15;

    const int b   = blockIdx.x;
    const int mt1 = b;
    const int mt2 = (CL_MTILES - 1) - b;
    const int n1  = CL_MTILES - b;       // #tiles in strip 1 (nt = mt1..511)

    // Stage both A tiles (each 16 x 512 f32) into LDS with 128-bit transfers.
    {
        const v4f* s1 = reinterpret_cast<const v4f*>(X + (size_t)mt1 * 16 * CL_D);
        const v4f* s2 = reinterpret_cast<const v4f*>(X + (size_t)mt2 * 16 * CL_D);
        v4f* d1 = reinterpret_cast<v4f*>(As);
        v4f* d2 = reinterpret_cast<v4f*>(As + 16 * CL_D);
#pragma unroll
        for (int i = 0; i < 8; ++i) {
            d1[tid + i * 256] = s1[tid + i * 256];
            d2[tid + i * 256] = s2[tid + i * 256];
        }
    }
    if (tid < 16)       Tg[tid]      = T[mt1 * 16 + tid];
    else if (tid < 32)  Tg[tid]      = T[mt2 * 16 + (tid - 16)];
    __syncthreads();

    float acc = 0.0f;

    for (int t = wave; t < CL_TOTAL; t += 8) {            // wave-uniform loop
        const bool second = (t >= n1);
        const int  mt     = second ? mt2 : mt1;
        const int  nt     = second ? (mt2 + (t - n1)) : (mt1 + t);
        const float w     = (nt == mt) ? 1.0f : 2.0f;     // symmetry weight

        const int    n    = nt * 16 + l16;                // global column
        // ISA 16x4 f32 A layout: lanes 0-15 -> K=ks+0,ks+1 ; lanes 16-31 -> K=ks+2,ks+3
        const float* Arow = As + (second ? 16 * CL_D : 0) + l16 * CL_D + 2 * half;
        const float* Brow = X + (size_t)n * CL_D + 2 * half;
        const int    tcol = T[n];                         // hot in L2

        // Hint the next tile's B rows toward the caches.
        if (t + 8 < CL_TOTAL) {
            const bool s2   = (t + 8 >= n1);
            const int  ntn  = s2 ? (mt2 + (t + 8 - n1)) : (mt1 + t + 8);
            __builtin_prefetch(X + (size_t)(ntn * 16 + l16) * CL_D, 0, 0);
        }

        v8f c = {};
#pragma unroll 8
        for (int ks = 0; ks < CL_D; ks += 4) {
            v2f av = *reinterpret_cast<const v2f*>(Arow + ks);   // ds_load_b64
            v2f bv = *reinterpret_cast<const v2f*>(Brow + ks);   // global_load_b64
            // 8 args: (neg_a, A, neg_b, B, c_mod, C, reuse_a, reuse_b)
            c = __builtin_amdgcn_wmma_f32_16x16x4_f32(
                    false, av, false, bv, (short)0, c, false, false);
        }

        // Branchless masked loss.  C/D layout: elem j -> row mt*16 + j + 8*half.
        const int tbase = (second ? 16 : 0) + 8 * half;
#pragma unroll
        for (int j = 0; j < 8; ++j) {
            const float s    = c[j];
            const bool  same = (Tg[tbase + j] == tcol);
            const float pos  = (s < 1.0f)      ? (1.0f - s) : 0.0f;
            const float neg  = (s > CL_MARGIN) ? s          : 0.0f;
            acc = fmaf(w, same ? pos : neg, acc);          // cndmask + fma
        }
    }

    // Deterministic fixed-order block reduction.
    Red[tid] = acc;
    __syncthreads();
#pragma unroll
    for (int off = 128; off > 0; off >>= 1) {
        if (tid < off) Red[tid] += Red[tid + off];
        __syncthreads();
    }
    if (tid == 0) partial[blockIdx.x] = Red[0];
}

// ---------------------------------------------------------------------------
// Kernel 2: deterministic reduction of the 256 per-block partials -> sum / N.
// ---------------------------------------------------------------------------
__global__ __launch_bounds__(256) void
contrastive_reduce_kernel(const float* __restrict__ partial,
                          float*       __restrict__ out)
{
    __shared__ float Red[256];
    const int tid = threadIdx.x;
    Red[tid] = partial[tid];
    __syncthreads();
#pragma unroll
    for (int off = 128; off > 0; off >>= 1) {
        if (tid < off) Red[tid] += Red[tid + off];
        __syncthreads();
    }
    if (tid == 0) out[0] = Red[0] * (1.0f / (float)CL_N);
}

// ---------------------------------------------------------------------------
extern "C" void kernel_launch(void* const* d_in, const int* in_sizes, int n_in,
                              void* d_out, int out_size, void* d_ws, size_t ws_size,
                              hipStream_t stream)
{
    (void)in_sizes; (void)n_in; (void)out_size; (void)ws_size;

    const float* X = (const float*)d_in[0];   // [8192, 512] fp32
    const int*   T = (const int*)d_in[1];     // [8192] labels
    float* out     = (float*)d_out;           // scalar
    float* partial = (float*)d_ws;            // 256 floats of scratch

    const size_t lds_bytes = 2u * 16u * CL_D * sizeof(float);   // 64 KB
    contrastive_tile_kernel<<<CL_BLOCKS, 256, lds_bytes, stream>>>(X, T, partial);
    contrastive_reduce_kernel<<<1, 256, 0, stream>>>(partial, out);
}